// involution_28166395528069
// MI455X (gfx1250) — compile-verified
//
#include <hip/hip_runtime.h>

typedef __attribute__((ext_vector_type(2))) float v2f;
typedef __attribute__((ext_vector_type(8))) float v8f;
typedef __attribute__((ext_vector_type(4))) int s4i;
typedef __attribute__((ext_vector_type(8))) int s8i;

#define BN_EPS 1e-5f

constexpr int B_ = 4, C_ = 256, H_ = 64, W_ = 64, HW = H_ * W_;
constexpr int CM = 64;                 // C / reduction
constexpr int G_ = 16, GC_ = 16, K2 = 49;

// D = A(16x4 f32) * B(4x16 f32) + C(16x16 f32), exact fp32 path on CDNA5
__device__ __forceinline__ v8f wmma_f32x4(v2f a, v2f b, v8f c) {
  return __builtin_amdgcn_wmma_f32_16x16x4_f32(false, a, false, b, (short)0, c,
                                               false, false);
}

// ---------------------------------------------------------------------------
// Tensor Data Mover: 2D fp32 tile (tile_d0 contiguous elems x tile_d1 rows,
// row stride stride_elems) global -> LDS. Descriptor per cdna5 D# layout:
//   group0: count=1 | lds_addr | global_addr[56:0] | type=2
//   group1: data_size=4B | tensor_dim0/1 | tile_dim0/1 | tensor_dim0_stride
// Issue from ONE wave only; completion tracked on TENSORcnt.
// ---------------------------------------------------------------------------
__device__ __forceinline__ void tdm_load_2d_f32(unsigned lds_off,
                                                const float* gptr,
                                                unsigned tile_d0,
                                                unsigned tile_d1,
                                                unsigned tensor_d0,
                                                unsigned tensor_d1,
                                                unsigned stride_elems) {
  unsigned long long ga = (unsigned long long)(size_t)gptr;
  s4i g0;
  g0[0] = 1;                                    // count=1, user descriptor
  g0[1] = (int)lds_off;                         // lds_addr (bytes)
  g0[2] = (int)(unsigned)ga;                    // global_addr[31:0]
  g0[3] = (int)(((unsigned)(ga >> 32) & 0x1FFFFFFu) | (2u << 30));  // [56:32]|type=2
  s8i g1;
  g1[0] = (int)(2u << 16);                      // data_size = 4 bytes, mask=0
  g1[1] = (int)((tensor_d0 & 0xFFFFu) << 16);   // dim0[15:0]
  g1[2] = (int)(((tensor_d0 >> 16) & 0xFFFFu) | ((tensor_d1 & 0xFFFFu) << 16));
  g1[3] = (int)(((tensor_d1 >> 16) & 0xFFFFu) | (tile_d0 << 16));
  g1[4] = (int)(tile_d1 & 0xFFFFu);             // tile_dim1, tile_dim2=0
  g1[5] = (int)stride_elems;                    // tensor_dim0_stride[31:0]
  g1[6] = 0;                                    // stride[47:32], dim1_stride lo
  g1[7] = 0;
  asm volatile("tensor_load_to_lds %0, %1" ::"s"(g0), "s"(g1) : "memory");
}

// ---------------------------------------------------------------------------
// Kernel 1: t[b][m][hw] = relu(bn1(w1[64x256] @ x[b][256][hw]))
// grid: 128 WGs (4 batches x 32 col-groups of 128), block: 256 (8 waves)
// each wave: one 16-col strip, 4 M-tiles of 16, K=256 in steps of 4
// ---------------------------------------------------------------------------
__global__ __launch_bounds__(256) void gemm1_bn_relu(
    const float* __restrict__ x, const float* __restrict__ w1,
    const float* __restrict__ g1, const float* __restrict__ be1,
    const float* __restrict__ mu1, const float* __restrict__ va1,
    float* __restrict__ t) {
  __shared__ float w1Lds[CM * C_];      // 64 KB, row-major [64][256]
  __shared__ float invLds[CM];
  __shared__ float biasLds[CM];

  const int tid = threadIdx.x;
  const int lane = tid & 31, wave = tid >> 5;
  const int half = lane >> 4, lm = lane & 15;
  const int b = blockIdx.x >> 5;
  const int hw0 = (blockIdx.x & 31) * 128;

  // TDM: stage all of w1 (64x256 fp32, contiguous) as one tensor DMA
  if (tid < 32) {
    tdm_load_2d_f32((unsigned)(size_t)&w1Lds[0], w1,
                    /*tile*/ C_, CM, /*tensor*/ C_, CM, /*stride*/ C_);
  }
  if (tid < CM) {
    float inv = g1[tid] * rsqrtf(va1[tid] + BN_EPS);
    invLds[tid] = inv;
    biasLds[tid] = be1[tid] - mu1[tid] * inv;
  }
  if (tid < 32) __builtin_amdgcn_s_wait_tensorcnt(0);
  __syncthreads();

  const int n0 = hw0 + wave * 16;
  const float* xb = x + (size_t)b * C_ * HW;

  v8f acc0 = {}, acc1 = {}, acc2 = {}, acc3 = {};
  for (int kk = 0; kk < C_; kk += 4) {
    const int krow = kk + 2 * half;  // lanes 0-15 -> K=kk,kk+1; 16-31 -> kk+2,kk+3
    v2f bf;
    bf[0] = xb[krow * HW + n0 + lm];
    bf[1] = xb[(krow + 1) * HW + n0 + lm];
    v2f a0 = *(const v2f*)&w1Lds[(0 * 16 + lm) * C_ + krow];
    v2f a1 = *(const v2f*)&w1Lds[(1 * 16 + lm) * C_ + krow];
    v2f a2 = *(const v2f*)&w1Lds[(2 * 16 + lm) * C_ + krow];
    v2f a3 = *(const v2f*)&w1Lds[(3 * 16 + lm) * C_ + krow];
    acc0 = wmma_f32x4(a0, bf, acc0);
    acc1 = wmma_f32x4(a1, bf, acc1);
    acc2 = wmma_f32x4(a2, bf, acc2);
    acc3 = wmma_f32x4(a3, bf, acc3);
  }

  float* tb = t + (size_t)b * CM * HW;
  v8f accs[4] = {acc0, acc1, acc2, acc3};
#pragma unroll
  for (int mt = 0; mt < 4; ++mt)
#pragma unroll
    for (int i = 0; i < 8; ++i) {
      const int row = mt * 16 + i + 8 * half;  // C/D layout
      float v = accs[mt][i] * invLds[row] + biasLds[row];
      tb[row * HW + n0 + lm] = fmaxf(v, 0.f);
    }
}

// ---------------------------------------------------------------------------
// Kernel 2: fused wgt-GEMM + involution aggregation + bn2 + relu
// grid: B*H = 256 WGs, block: 256 (8 waves). One image row per WG.
// per group g: wgt_g[49x64] = w2_g[49x64] @ tT[64x64] + b2 (WMMA, LDS-resident)
//              out[c][w] = relu(bn2( sum_k xPatch[c][k] * wgt_g[k][w] ))
// ---------------------------------------------------------------------------
__global__ __launch_bounds__(256) void invol_fused(
    const float* __restrict__ x, const float* __restrict__ t,
    const float* __restrict__ w2, const float* __restrict__ b2,
    const float* __restrict__ g2, const float* __restrict__ be2,
    const float* __restrict__ mu2, const float* __restrict__ va2,
    float* __restrict__ out) {
  __shared__ float tT[CM * 64];            // 16 KB   [k][n]
  __shared__ float wgtLds[K2 * 64];        // 12.25KB [k2][w]
  __shared__ float xPatch[GC_ * 7 * 72];   // 31.5 KB [gc][ky][col], stride 72
  __shared__ float inv2Lds[C_];
  __shared__ float bias2Lds[C_];

  const int tid = threadIdx.x;
  const int lane = tid & 31, wave = tid >> 5;
  const int half = lane >> 4, lm = lane & 15;
  const int b = blockIdx.x >> 6;
  const int h = blockIdx.x & 63;

  // TDM: stage t tile (64 rows(k) x 64 cols(hw), row stride HW) into LDS
  const float* tb = t + (size_t)b * CM * HW + h * 64;
  if (tid < 32) {
    tdm_load_2d_f32((unsigned)(size_t)&tT[0], tb,
                    /*tile*/ 64, CM, /*tensor*/ HW, CM, /*stride*/ HW);
  }
  if (tid < C_) {
    float inv = g2[tid] * rsqrtf(va2[tid] + BN_EPS);
    inv2Lds[tid] = inv;
    bias2Lds[tid] = be2[tid] - mu2[tid] * inv;
  }
  if (tid < 32) __builtin_amdgcn_s_wait_tensorcnt(0);
  __syncthreads();

  for (int g = 0; g < G_; ++g) {
    // stage zero-padded x patch: 16 ch x 7 rows x 70 cols (manual: needs
    // zero fill at negative offsets, which TDM OOB handling can't express)
    for (int i = tid; i < GC_ * 7 * 70; i += 256) {
      const int gc = i / 490, rem = i % 490;
      const int r = rem / 70, col = rem % 70;
      const int y = h - 3 + r, xw = col - 3;
      float v = 0.f;
      if ((unsigned)y < 64u && (unsigned)xw < 64u)
        v = x[(((size_t)b * C_ + g * GC_ + gc) * H_ + y) * W_ + xw];
      xPatch[gc * 504 + r * 72 + col] = v;
    }

    // wgt GEMM: 4 M-tiles (49 rows padded to 64) x 4 N-strips, 2 tiles/wave
    for (int tile = wave; tile < 16; tile += 8) {
      const int mt = tile >> 2, n0 = (tile & 3) * 16;
      const int m = mt * 16 + lm;
      const int grow = g * K2 + (m < K2 ? m : K2 - 1);  // clamp pad rows
      const float* w2row = w2 + grow * CM;
      v8f acc = {};
      for (int kk = 0; kk < CM; kk += 4) {
        const int krow = kk + 2 * half;
        v2f af = *(const v2f*)&w2row[krow];
        v2f bf;
        bf[0] = tT[krow * 64 + n0 + lm];
        bf[1] = tT[(krow + 1) * 64 + n0 + lm];
        acc = wmma_f32x4(af, bf, acc);
      }
#pragma unroll
      for (int i = 0; i < 8; ++i) {
        const int mr = mt * 16 + i + 8 * half;
        if (mr < K2) wgtLds[mr * 64 + n0 + lm] = acc[i] + b2[g * K2 + mr];
      }
    }
    __syncthreads();

    // aggregation: 16 channels x 64 pixels, 4 outputs/thread
    const int w = tid & 63;
    for (int it = 0; it < 4; ++it) {
      const int cl = it * 4 + (tid >> 6);
      const float* xp = &xPatch[cl * 504 + w];
      float s = 0.f;
#pragma unroll
      for (int ki = 0; ki < 7; ++ki)
#pragma unroll
        for (int kj = 0; kj < 7; ++kj)
          s += xp[ki * 72 + kj] * wgtLds[(ki * 7 + kj) * 64 + w];
      const int c = g * GC_ + cl;
      float v = s * inv2Lds[c] + bias2Lds[c];
      out[(((size_t)b * C_ + c) * H_ + h) * W_ + w] = fmaxf(v, 0.f);
    }
    __syncthreads();
  }
}

extern "C" void kernel_launch(void* const* d_in, const int* in_sizes, int n_in,
                              void* d_out, int out_size, void* d_ws,
                              size_t ws_size, hipStream_t stream) {
  (void)in_sizes; (void)n_in; (void)out_size; (void)ws_size;
  const float* x   = (const float*)d_in[0];
  const float* w1  = (const float*)d_in[1];
  const float* g1  = (const float*)d_in[2];
  const float* be1 = (const float*)d_in[3];
  const float* mu1 = (const float*)d_in[4];
  const float* va1 = (const float*)d_in[5];
  const float* w2  = (const float*)d_in[6];
  const float* b2  = (const float*)d_in[7];
  const float* g2  = (const float*)d_in[8];
  const float* be2 = (const float*)d_in[9];
  const float* mu2 = (const float*)d_in[10];
  const float* va2 = (const float*)d_in[11];
  float* out = (float*)d_out;
  float* t = (float*)d_ws;  // 4 * 64 * 4096 floats = 4 MB scratch

  gemm1_bn_relu<<<128, 256, 0, stream>>>(x, w1, g1, be1, mu1, va1, t);
  invol_fused<<<B_ * H_, 256, 0, stream>>>(x, t, w2, b2, g2, be2, mu2, va2, out);
}